// EdgewiseReduce_64776696758368
// MI455X (gfx1250) — compile-verified
//
#include <hip/hip_runtime.h>
#include <stdint.h>

// EdgewiseReduce: out[n, :] = sum_{e : dst[e]==n} edge_data[e, :] * 0.5
// E = in_sizes[1] edges, D = 32 features, out_size = N*32.
//
// Memory-bound scatter (~220 MB traffic -> ~9.5us at 23.3 TB/s). Output
// (12.8 MB) lives in the 192 MB L2, so f32 atomics resolve L2-side.
// CDNA5 path used: async global->LDS (ASYNCcnt) double-buffered pipeline
// per wave, decoupling HBM streaming from atomic issue. Index broadcast
// uses v_readlane_b32 (imm lane) instead of ds_bpermute to keep LDS
// bandwidth free for the tile reads.

#define D            32
#define TILE_EDGES   32            // one tile = 32 edges x 32 feats = 4 KB
#define WAVES        8             // 256 threads / block
#define FACTOR       0.5f

__global__ void zero_f4(float4* __restrict__ out, int n4) {
  int i = blockIdx.x * blockDim.x + threadIdx.x;
  if (i < n4) out[i] = make_float4(0.f, 0.f, 0.f, 0.f);
}

// Wave-cooperative async copy of one 4 KB tile: 8 chunks of (32 lanes x 16B).
// VDST operand = LDS byte address (low 32 bits of the generic shared ptr),
// VADDR = full 64-bit global address (GV mode).
__device__ __forceinline__ void async_tile_load(const float* __restrict__ g,
                                                const float* lbase, int lane) {
  uint32_t laddr = (uint32_t)(uintptr_t)lbase + (uint32_t)lane * 16u;
  const char* gaddr = (const char*)g + (size_t)lane * 16u;
#pragma unroll
  for (int c = 0; c < 8; ++c) {
    asm volatile("global_load_async_to_lds_b128 %0, %1, off"
                 :: "v"(laddr + (uint32_t)c * 512u), "v"(gaddr + c * 512)
                 : "memory");
  }
}

__global__ __launch_bounds__(WAVES * 32) void edgewise_scatter(
    const float* __restrict__ edge_data, const int* __restrict__ edge_dst,
    float* __restrict__ out, int ntiles) {
  // 8 waves x double buffer x 4 KB = 64 KB LDS (WGP has 320 KB).
  __shared__ float sbuf[WAVES][2][TILE_EDGES * D];

  const int lane = threadIdx.x & 31;
  const int wid  = threadIdx.x >> 5;
  const int gw   = blockIdx.x * WAVES + wid;   // global wave id
  const int nw   = gridDim.x * WAVES;          // total waves (tile stride)

  int tile = gw;
  if (tile >= ntiles) return;

  // Prologue: prefetch first tile into buffer 0.
  async_tile_load(edge_data + (size_t)tile * (TILE_EDGES * D),
                  &sbuf[wid][0][0], lane);
  int dst_cur = edge_dst[(size_t)tile * TILE_EDGES + lane];
  int b = 0;

  for (;;) {
    const int next = tile + nw;
    int dst_next = 0;
    if (next < ntiles) {
      // Issue prefetch for next tile, then wait until only those 8 async
      // ops are outstanding => current tile's data has landed in LDS.
      async_tile_load(edge_data + (size_t)next * (TILE_EDGES * D),
                      &sbuf[wid][b ^ 1][0], lane);
      dst_next = edge_dst[(size_t)next * TILE_EDGES + lane];
      asm volatile("s_wait_asynccnt 8" ::: "memory");
    } else {
      asm volatile("s_wait_asynccnt 0" ::: "memory");
    }

    // Compute phase: lane l owns feature l of every edge in the tile.
    // Fully unrolled -> readlane with immediate lane index (no LDS traffic);
    // tile reads are bank-conflict-free ds_load_b32 (consecutive 4B across
    // 32 lanes over 64 banks).
    const float* buf = &sbuf[wid][b][0];
#pragma unroll
    for (int e = 0; e < TILE_EDGES; ++e) {
      int d   = __builtin_amdgcn_readlane(dst_cur, e);  // v_readlane_b32 imm
      float v = buf[e * D + lane] * FACTOR;
      atomicAdd(out + (size_t)d * D + lane, v);  // global_atomic_add_f32, L2-resident
    }

    if (next >= ntiles) break;
    tile = next;
    b ^= 1;
    dst_cur = dst_next;
  }
}

// Tail for E % 32 edges (zero for the reference shapes, kept for robustness).
__global__ void edgewise_tail(const float* __restrict__ edge_data,
                              const int* __restrict__ edge_dst,
                              float* __restrict__ out, int start, int E) {
  int idx = blockIdx.x * blockDim.x + threadIdx.x;
  int e = start + (idx >> 5);
  if (e < E) {
    int f = idx & 31;
    atomicAdd(out + (size_t)edge_dst[e] * D + f,
              edge_data[(size_t)e * D + f] * FACTOR);
  }
}

extern "C" void kernel_launch(void* const* d_in, const int* in_sizes, int n_in,
                              void* d_out, int out_size, void* d_ws, size_t ws_size,
                              hipStream_t stream) {
  const float* edge_data = (const float*)d_in[0];
  const int*   edge_dst  = (const int*)d_in[1];   // harness: integer -> const int*
  float*       out       = (float*)d_out;
  const int    E         = in_sizes[1];

  // 1) Zero the (poisoned) output: N*32 floats, vectorized float4 stores.
  const int n4 = out_size / 4;
  zero_f4<<<(n4 + 255) / 256, 256, 0, stream>>>((float4*)d_out, n4);

  // 2) Pipelined scatter over full 32-edge tiles.
  const int ntiles = E / TILE_EDGES;
  if (ntiles > 0) {
    int blocks = (ntiles + WAVES - 1) / WAVES;
    if (blocks > 4096) blocks = 4096;   // more waves in flight for atomic issue
    edgewise_scatter<<<blocks, WAVES * 32, 0, stream>>>(edge_data, edge_dst,
                                                        out, ntiles);
  }

  // 3) Remainder edges (simple per-feature atomics).
  const int rem = E - ntiles * TILE_EDGES;
  if (rem > 0) {
    const int threads = rem * D;
    edgewise_tail<<<(threads + 255) / 256, 256, 0, stream>>>(
        edge_data, edge_dst, out, ntiles * TILE_EDGES, E);
  }
}